// kernel_generated_3_62904091017420
// MI455X (gfx1250) — compile-verified
//
#include <hip/hip_runtime.h>

typedef float v2f __attribute__((ext_vector_type(2)));
typedef float v8f __attribute__((ext_vector_type(8)));

#define BB    128
#define GG    2
#define CIN   24
#define COUT  96
#define KW    7
#define HH    28
#define WW    28
#define KDIM  (CIN*KW)    // 168
#define XSTR  40          // padded LDS row stride (28 + 6 halo + pad)
#define NSTEP (KDIM/4)    // 42 wmma K-steps

// ---- prep: Apack[kg*96 + o] = w[i][o][k], kg = i*7 + k  (168 x 96 f32) ----
__global__ __launch_bounds__(256)
void pack_weights(const float* __restrict__ w, float* __restrict__ apack) {
    int t = blockIdx.x * 256 + threadIdx.x;
    if (t < KDIM * COUT) {
        int kg = t / COUT;
        int o  = t - kg * COUT;
        int i  = kg / KW;
        int k  = kg - i * KW;
        apack[t] = w[i * (COUT * KW) + o * KW + k];
    }
}

// ---- main: grouped conv1d as im2col GEMM via V_WMMA_F32_16X16X4_F32 ----
__global__ __launch_bounds__(128)
void conv_wmma(const float* __restrict__ x,
               const float* __restrict__ apack,
               float* __restrict__ out) {
    __shared__ float ldsX[GG * CIN * XSTR];     // 1920 floats = 7.5 KB

    const int b   = blockIdx.x;
    const int h   = blockIdx.y;
    const int tid = threadIdx.x;

    // zero LDS (covers halos + tail padding) : 1920 = 15 * 128
    #pragma unroll
    for (int j = 0; j < 15; ++j) ldsX[tid + j * 128] = 0.0f;
    __syncthreads();

    // stage 48 x-rows of width 28 at offset +3 (zero halo each side)
    for (int t = tid; t < GG * CIN * WW; t += 128) {
        int c  = t / WW;
        int wp = t - c * WW;
        ldsX[c * XSTR + 3 + wp] = x[((b * (GG * CIN) + c) * HH + h) * WW + wp];
    }
    __syncthreads();

    const int wid  = tid >> 5;         // 4 waves
    const int lane = tid & 31;
    const int half = lane >> 4;        // A/B K-slot half
    const int ln   = lane & 15;
    const int g    = wid >> 1;         // group 0/1
    const int mtb  = (wid & 1) * 3;    // M-tile base: {0..2} or {3..5}

    const float* xrow = ldsX + g * CIN * XSTR;

    v8f acc[3][2] = {};                // 3 M-tiles x 2 N-tiles, f32 16x16 each

    #pragma unroll
    for (int s = 0; s < NSTEP; ++s) {
        // compile-time im2col offsets (dwords) for the 4 K-values of this step
        const int c0 = ((4*s + 0) / KW) * XSTR + ((4*s + 0) % KW);
        const int c1 = ((4*s + 1) / KW) * XSTR + ((4*s + 1) % KW);
        const int c2 = ((4*s + 2) / KW) * XSTR + ((4*s + 2) % KW);
        const int c3 = ((4*s + 3) / KW) * XSTR + ((4*s + 3) % KW);
        const int off0 = half ? c2 : c0;
        const int off1 = half ? c3 : c1;

        // B fragments from LDS (halo-padded: branch-free)
        v2f bf[2];
        #pragma unroll
        for (int nt = 0; nt < 2; ++nt) {
            const int wp = nt * 16 + ln;
            bf[nt].x = xrow[off0 + wp];
            bf[nt].y = xrow[off1 + wp];
        }

        // A fragments from packed weights (L1-resident, coalesced 64B runs)
        const int arow = 4 * s + (half ? 2 : 0);
        #pragma unroll
        for (int m = 0; m < 3; ++m) {
            const int o = (mtb + m) * 16 + ln;
            v2f af;
            af.x = apack[arow * COUT + o];
            af.y = apack[(arow + 1) * COUT + o];
            #pragma unroll
            for (int nt = 0; nt < 2; ++nt) {
                acc[m][nt] = __builtin_amdgcn_wmma_f32_16x16x4_f32(
                    false, af, false, bf[nt],
                    (short)0, acc[m][nt], false, false);
            }
        }
    }

    // epilogue: roll along H by +1, streaming (non-temporal) stores
    const int hd = (h + 1 == HH) ? 0 : (h + 1);
    #pragma unroll
    for (int m = 0; m < 3; ++m) {
        #pragma unroll
        for (int nt = 0; nt < 2; ++nt) {
            const int wp = nt * 16 + ln;
            if (wp < WW) {
                #pragma unroll
                for (int v = 0; v < 8; ++v) {
                    const int o = (mtb + m) * 16 + half * 8 + v;   // D: M = v + 8*half
                    __builtin_nontemporal_store(
                        acc[m][nt][v],
                        &out[((b * (GG * COUT) + g * COUT + o) * HH + hd) * WW + wp]);
                }
            }
        }
    }
}

extern "C" void kernel_launch(void* const* d_in, const int* in_sizes, int n_in,
                              void* d_out, int out_size, void* d_ws, size_t ws_size,
                              hipStream_t stream) {
    const float* x  = (const float*)d_in[0];   // (128, 48, 28, 28) f32
    const float* w  = (const float*)d_in[1];   // (24, 96, 7) f32
    float* out      = (float*)d_out;           // (128, 192, 28, 28) f32
    float* apack    = (float*)d_ws;            // 168*96 f32 = 64.5 KB scratch

    pack_weights<<<(KDIM * COUT + 255) / 256, 256, 0, stream>>>(w, apack);

    dim3 grid(BB, HH);
    conv_wmma<<<grid, 128, 0, stream>>>(x, apack, out);
}